// CurriculumMining_40767829574564
// MI455X (gfx1250) — compile-verified
//
#include <hip/hip_runtime.h>

// Problem constants (reference: B=8192, D=128, f32)
#define BN 8192
#define DN 128

typedef float v2f __attribute__((ext_vector_type(2)));
typedef float v8f __attribute__((ext_vector_type(8)));

// ---------------------------------------------------------------------------
// Stage 2: per-row squared norms. One wave (32 lanes) per row, 4 elems/lane.
// ---------------------------------------------------------------------------
__global__ void row_norms_kernel(const float* __restrict__ X, float* __restrict__ out) {
    const int row  = blockIdx.x * 8 + (threadIdx.x >> 5);
    const int lane = threadIdx.x & 31;
    const float* p = X + (size_t)row * DN;
    float s = 0.0f;
    #pragma unroll
    for (int k = lane; k < DN; k += 32) {
        float v = p[k];
        s += v * v;
    }
    #pragma unroll
    for (int off = 16; off > 0; off >>= 1)
        s += __shfl_xor(s, off, 32);
    if (lane == 0) out[row] = s;
}

// ---------------------------------------------------------------------------
// Stage 3: rev[i,j] = t2[i] + s2[j] - 2 * (T @ S^T)[i,j], diag forced to -1.
// V_WMMA_F32_16X16X4_F32, K = 128 (32 steps).
// Register blocking: each wave produces a 32x64 strip = 2 (M) x 4 (N) tiles,
// so each K-step issues 2 A loads + 4 B loads feeding 8 WMMAs (0.75 loads
// per WMMA) with 8 independent accumulation chains.
// Block = 8 waves in a 2x4 arrangement -> 64 M x 256 N per block.
// ISA fragment layout (wave32):
//   A 16x4 : lane L holds A[M=L&15][K = 2*(L>>4) + {0,1}] in VGPR {0,1}
//   B 4x16 : lane L holds B[K = 2*(L>>4) + {0,1}][N = L&15]  (B = S^T tile,
//            so B[k][n] = S[n0+n][k] -> same row-major access pattern as A)
//   C/D    : VGPR r, lane L -> (M = r + 8*(L>>4), N = L&15)
// ---------------------------------------------------------------------------
__global__ void dist_gemm_kernel(const float* __restrict__ T,
                                 const float* __restrict__ S,
                                 const float* __restrict__ t2,
                                 const float* __restrict__ s2,
                                 float* __restrict__ rev) {
    const int wave = threadIdx.x >> 5;                // 0..7
    const int wm   = wave >> 2;                       // 0..1  (M group)
    const int wn   = wave & 3;                        // 0..3  (N group)
    const int lane = threadIdx.x & 31;
    const int half = lane >> 4;                       // 0 or 1
    const int l16  = lane & 15;

    const int m0 = (blockIdx.y * 2 + wm) * 32;        // wave's 32-row M strip
    const int n0 = (blockIdx.x * 4 + wn) * 64;        // wave's 64-col N strip

    const float* arow0 = T + (size_t)(m0 + l16) * DN + half * 2;
    const float* arow1 = arow0 + (size_t)16 * DN;
    const float* brow0 = S + (size_t)(n0 + l16) * DN + half * 2;
    const float* brow1 = brow0 + (size_t)16 * DN;
    const float* brow2 = brow0 + (size_t)32 * DN;
    const float* brow3 = brow0 + (size_t)48 * DN;

    v8f acc[2][4] = {};
    #pragma unroll
    for (int k = 0; k < DN; k += 4) {
        v2f a0 = *(const v2f*)(arow0 + k);            // 8B aligned
        v2f a1 = *(const v2f*)(arow1 + k);
        v2f b0 = *(const v2f*)(brow0 + k);
        v2f b1 = *(const v2f*)(brow1 + k);
        v2f b2 = *(const v2f*)(brow2 + k);
        v2f b3 = *(const v2f*)(brow3 + k);
        acc[0][0] = __builtin_amdgcn_wmma_f32_16x16x4_f32(false, a0, false, b0, (short)0, acc[0][0], false, false);
        acc[0][1] = __builtin_amdgcn_wmma_f32_16x16x4_f32(false, a0, false, b1, (short)0, acc[0][1], false, false);
        acc[0][2] = __builtin_amdgcn_wmma_f32_16x16x4_f32(false, a0, false, b2, (short)0, acc[0][2], false, false);
        acc[0][3] = __builtin_amdgcn_wmma_f32_16x16x4_f32(false, a0, false, b3, (short)0, acc[0][3], false, false);
        acc[1][0] = __builtin_amdgcn_wmma_f32_16x16x4_f32(false, a1, false, b0, (short)0, acc[1][0], false, false);
        acc[1][1] = __builtin_amdgcn_wmma_f32_16x16x4_f32(false, a1, false, b1, (short)0, acc[1][1], false, false);
        acc[1][2] = __builtin_amdgcn_wmma_f32_16x16x4_f32(false, a1, false, b2, (short)0, acc[1][2], false, false);
        acc[1][3] = __builtin_amdgcn_wmma_f32_16x16x4_f32(false, a1, false, b3, (short)0, acc[1][3], false, false);
    }

    // Epilogue: hoist the 16 t2 values for this lane's M rows, then emit.
    float tm[2][8];
    #pragma unroll
    for (int i = 0; i < 2; ++i)
        #pragma unroll
        for (int r = 0; r < 8; ++r)
            tm[i][r] = t2[m0 + i * 16 + r + 8 * half];

    #pragma unroll
    for (int j = 0; j < 4; ++j) {
        const int   n   = n0 + j * 16 + l16;
        const float s2n = s2[n];
        #pragma unroll
        for (int i = 0; i < 2; ++i) {
            #pragma unroll
            for (int r = 0; r < 8; ++r) {
                const int m = m0 + i * 16 + r + 8 * half;
                float val = tm[i][r] + s2n - 2.0f * acc[i][j][r];
                if (m == n) val = -1.0f;              // diag override
                rev[(size_t)m * BN + n] = val;
            }
        }
    }
}

// ---------------------------------------------------------------------------
// Stage 4: per-row rank selection + gather.
// One block (256 threads) per row. Row keys cached once in LDS (32KB of the
// 320KB WGP LDS), then:
//   - 4x MSB-first radix histogram passes -> threshold value v at ascending
//     rank (B-1-kd)
//   - count G = #(u > v)
//   - stable pick of the (kd - G)-th tied index (matches stable argsort)
//   - gather student[negIdx] row into the output
// ---------------------------------------------------------------------------
__global__ void select_gather_kernel(const float* __restrict__ rev,
                                     const float* __restrict__ S,
                                     const int* __restrict__ tau_pct,
                                     float* __restrict__ out_neg) {
    __shared__ unsigned keys[BN];      // 32 KB
    __shared__ unsigned hist[256];     // histogram / per-thread counts (reused)
    __shared__ int s_bucket, s_rem, s_G, s_idx;

    const int row = blockIdx.x;
    const int tid = threadIdx.x;
    const float* rp = rev + (size_t)row * BN;

    // idx_threshold, on device (round-half-even like Python round())
    const float tau = (float)tau_pct[0] * 0.01f;
    int kd = (BN - 1) - __float2int_rn(tau * (float)(BN - 1));
    if (kd == BN - 1) kd = BN - 2;
    const int ka = (BN - 1) - kd;      // ascending rank of threshold value

    // Load row once; transform to order-preserving unsigned keys.
    for (int j = tid; j < BN; j += 256) {
        unsigned fb = __float_as_uint(rp[j]);
        keys[j] = fb ^ ((fb & 0x80000000u) ? 0xFFFFFFFFu : 0x80000000u);
    }
    __syncthreads();

    // MSB-first radix select of the value at ascending rank ka.
    unsigned prefixKey = 0, prefixMask = 0;
    int remaining = ka;
    for (int shift = 24; shift >= 0; shift -= 8) {
        hist[tid] = 0;
        __syncthreads();
        for (int j = tid; j < BN; j += 256) {
            unsigned u = keys[j];
            if ((u & prefixMask) == prefixKey)
                atomicAdd(&hist[(u >> shift) & 0xFFu], 1u);
        }
        __syncthreads();
        if (tid == 0) {
            int rem = remaining, bucket = 255;
            for (int i = 0; i < 256; ++i) {
                int c = (int)hist[i];
                if (rem < c) { bucket = i; break; }
                rem -= c;
            }
            s_bucket = bucket;
            s_rem = rem;
        }
        __syncthreads();
        prefixKey |= ((unsigned)s_bucket) << shift;
        prefixMask |= 0xFFu << shift;
        remaining = s_rem;
        __syncthreads();
    }
    const unsigned vkey = prefixKey;

    // G = count(u > vkey)  (these occupy descending ranks [0, G))
    if (tid == 0) { s_G = 0; s_idx = 0; }
    __syncthreads();
    {
        int local = 0;
        for (int j = tid; j < BN; j += 256)
            if (keys[j] > vkey) local++;
        atomicAdd(&s_G, local);
    }
    __syncthreads();
    const int r = kd - s_G;            // 0-indexed within the tied group

    // Stable pick: contiguous 32-element chunk per thread, exclusive prefix.
    const int base = tid * (BN / 256);
    {
        int local_eq = 0;
        #pragma unroll
        for (int j = 0; j < BN / 256; ++j)
            if (keys[base + j] == vkey) local_eq++;
        hist[tid] = (unsigned)local_eq;
    }
    __syncthreads();
    if (tid == 0) {
        unsigned s = 0;
        for (int i = 0; i < 256; ++i) { unsigned c = hist[i]; hist[i] = s; s += c; }
    }
    __syncthreads();
    {
        int rank = (int)hist[tid];
        #pragma unroll
        for (int j = 0; j < BN / 256; ++j) {
            if (keys[base + j] == vkey) {
                if (rank == r) s_idx = base + j;   // globally unique rank -> one writer
                rank++;
            }
        }
    }
    __syncthreads();

    // Gather student[negIdx] row (128 floats) into output.
    const int negIdx = s_idx;
    if (tid < DN)
        out_neg[(size_t)row * DN + tid] = S[(size_t)negIdx * DN + tid];
}

// ---------------------------------------------------------------------------
// Launcher
// ---------------------------------------------------------------------------
extern "C" void kernel_launch(void* const* d_in, const int* in_sizes, int n_in,
                              void* d_out, int out_size, void* d_ws, size_t ws_size,
                              hipStream_t stream) {
    (void)in_sizes; (void)n_in; (void)out_size; (void)ws_size;

    const float* T   = (const float*)d_in[0];
    const float* S   = (const float*)d_in[1];
    const int*   tau = (const int*)d_in[2];

    float* ws  = (float*)d_ws;
    float* rev = ws;                           // BN*BN floats (256 MB)
    float* t2  = ws + (size_t)BN * BN;         // BN floats
    float* s2  = t2 + BN;                      // BN floats

    float* out_teacher = (float*)d_out;
    float* out_neg     = out_teacher + (size_t)BN * DN;

    // Output #1: teacher passthrough.
    hipMemcpyAsync(out_teacher, T, (size_t)BN * DN * sizeof(float),
                   hipMemcpyDeviceToDevice, stream);

    // Squared norms.
    row_norms_kernel<<<BN / 8, 256, 0, stream>>>(T, t2);
    row_norms_kernel<<<BN / 8, 256, 0, stream>>>(S, s2);

    // Distance matrix via f32 WMMA.
    // Grid: x covers N in 256-col blocks, y covers M in 64-row blocks.
    dim3 ggrid(BN / 256, BN / 64);
    dist_gemm_kernel<<<ggrid, 256, 0, stream>>>(T, S, t2, s2, rev);

    // Per-row rank selection + gather (output #2).
    select_gather_kernel<<<BN, 256, 0, stream>>>(rev, S, tau, out_neg);
}